// NMSLayer_68796786147721
// MI455X (gfx1250) — compile-verified
//
#include <hip/hip_runtime.h>
#include <hip/hip_bf16.h>

// ---------------------------------------------------------------------------
// NMS for N=8192 boxes, MI455X (gfx1250, wave32).
//   d_in[0]: boxes  [N,4] f32 (x1,y1,x2,y2)
//   d_in[1]: scores [N]   f32
//   d_out  : keep mask [N] f32 (1.0 keep / 0.0 suppressed), original order
// Workspace layout (d_ws):
//   [0]                : order      N x u32      (sorted pos -> original idx)
//   [N*4]              : sboxes     N x float4   (boxes in sorted order)
//   [N*4 + N*16]       : sup        N x (N/32) u32  suppression bit matrix
// ---------------------------------------------------------------------------

#define NMS_N      8192
#define NMS_WORDS  (NMS_N / 32)   // 256 u32 words per row
#define IOU_THRESH 0.5f

// ---------------------------------------------------------------------------
// Kernel 1: single-block bitonic argsort (descending score, stable).
// Key = (~score_bits << 32) | index : ascending u64 sort == stable descending
// score sort (scores are non-negative floats so uint compare == float compare).
// 8192 x 8B = 64KB static LDS -- fits the WGP's 320KB LDS budget.
// ---------------------------------------------------------------------------
__global__ __launch_bounds__(1024) void nms_sort_kernel(
    const float4* __restrict__ boxes, const float* __restrict__ scores,
    unsigned* __restrict__ order, float4* __restrict__ sboxes, int n) {
  __shared__ unsigned long long skey[NMS_N];
  const int tid = threadIdx.x;
  const int T = 1024;

  for (int i = tid; i < NMS_N; i += T) {
    if (i < n) {
      unsigned sb = __float_as_uint(scores[i]);
      skey[i] = ((unsigned long long)(sb ^ 0xFFFFFFFFu) << 32) | (unsigned)i;
    } else {
      skey[i] = ~0ull;  // pad sinks to the end of ascending sort
    }
  }

  for (unsigned k = 2; k <= NMS_N; k <<= 1) {
    for (unsigned j = k >> 1; j > 0; j >>= 1) {
      __syncthreads();
      for (unsigned i = tid; i < NMS_N; i += T) {
        unsigned l = i ^ j;
        if (l > i) {
          unsigned long long a = skey[i], b = skey[l];
          bool up = ((i & k) == 0);
          if ((a > b) == up) { skey[i] = b; skey[l] = a; }
        }
      }
    }
  }
  __syncthreads();

  for (int i = tid; i < n; i += T) {
    unsigned idx = (unsigned)(skey[i] & 0xFFFFFFFFu);
    order[i] = idx;
    sboxes[i] = boxes[idx];
  }
}

// ---------------------------------------------------------------------------
// Kernel 2: suppression bit-matrix tiles (64 rows x 64 cols per block).
// Row/col boxes are staged into LDS with gfx1250 async global->LDS B128 loads
// (ASYNCcnt tracked), then 8 waves ballot-pack 32 IoU predicates per word.
// Every (row, word) of the matrix is written (scratch is poisoned by harness).
// ---------------------------------------------------------------------------
__global__ __launch_bounds__(256) void nms_sup_tile_kernel(
    const float4* __restrict__ sboxes, unsigned* __restrict__ sup, int n) {
  __shared__ float4 rbox[64];
  __shared__ float4 cbox[64];

  const int row0 = blockIdx.y * 64;
  const int col0 = blockIdx.x * 64;
  const int tid = threadIdx.x;

  // Waves 0-1 stage the 64 row boxes, waves 2-3 the 64 col boxes: one
  // async B128 (one box) per lane, landing directly in LDS.
  if (tid < 128) {
    const int which = tid >> 6;  // 0 = rows, 1 = cols
    const int l = tid & 63;
    unsigned long long gaddr =
        (unsigned long long)(sboxes + (which ? (col0 + l) : (row0 + l)));
    unsigned ldsoff = (unsigned)(unsigned long long)(
        which ? (const void*)&cbox[l] : (const void*)&rbox[l]);
    asm volatile("global_load_async_to_lds_b128 %0, %1, off"
                 :: "v"(ldsoff), "v"(gaddr) : "memory");
  }
  asm volatile("s_wait_asynccnt 0" ::: "memory");
  __syncthreads();

  const int wave = tid >> 5;
  const int lane = tid & 31;
  const int words = n >> 5;

  // 64 rows x 2 half-words = 128 wave-tasks; 8 waves -> 16 iterations.
  for (int it = 0; it < 16; ++it) {
    const int task = it * 8 + wave;
    const int r = task >> 1;       // local row 0..63
    const int h = task & 1;        // which 32-col half
    const int jl = h * 32 + lane;  // local col 0..63

    const float4 bi = rbox[r];
    const float4 bj = cbox[jl];
    const float ai = (bi.z - bi.x + 1.0f) * (bi.w - bi.y + 1.0f);
    const float aj = (bj.z - bj.x + 1.0f) * (bj.w - bj.y + 1.0f);
    const float xx1 = fmaxf(bi.x, bj.x);
    const float yy1 = fmaxf(bi.y, bj.y);
    const float xx2 = fminf(bi.z, bj.z);
    const float yy2 = fminf(bi.w, bj.w);
    const float w  = fmaxf(xx2 - xx1 + 1.0f, 0.0f);
    const float hh = fmaxf(yy2 - yy1 + 1.0f, 0.0f);
    const float inter = w * hh;
    const float iou = inter / (ai + aj - inter);

    const int gi = row0 + r;
    const int gj = col0 + jl;
    const bool s = (iou > IOU_THRESH) && (gj > gi);

    const unsigned m = __builtin_amdgcn_ballot_w32(s);  // wave32 ballot
    if (lane == 0) {
      sup[(size_t)gi * words + (unsigned)(col0 >> 5) + h] = m;
    }
  }
}

// ---------------------------------------------------------------------------
// Kernel 3: sequential greedy scan, single wave32, barrier-free.
// The full 8192-bit keep mask lives in 8 VGPRs per lane (word w owned by
// lane w&31, slot w>>5). Per row i:
//   - keep bit broadcast via v_readlane (slot i>>10 is uniform & constant
//     per unrolled outer block -> static register indexing, no LDS, no bar)
//   - if kept, 8 coalesced dword loads/lane apply ~row_i to the mask
//   - one global_prefetch_b8 (32B/lane stride) covers the 1KB row after next
// Then keep bits are scattered to original order.
// ---------------------------------------------------------------------------
__global__ __launch_bounds__(32) void nms_scan_kernel(
    const unsigned* __restrict__ sup, const unsigned* __restrict__ order,
    float* __restrict__ out, int n) {
  const int lane = threadIdx.x;  // single wave32

  unsigned kw[8];
#pragma unroll
  for (int t = 0; t < 8; ++t) kw[t] = 0xFFFFFFFFu;

#pragma unroll
  for (int s = 0; s < 8; ++s) {
    for (int ii = 0; ii < 1024; ++ii) {
      const int i = s * 1024 + ii;

      // Prefetch the row after next (whole 1KB row: 32 lanes x 32B stride).
      const int pi = (i + 2 < n) ? (i + 2) : i;
      __builtin_prefetch(&sup[(size_t)pi * NMS_WORDS + lane * 8], 0, 0);

      // Broadcast keep bit i from its owning lane (wave32 readlane).
      const unsigned w = __builtin_amdgcn_readlane(kw[s], (i >> 5) & 31);
      if ((w >> (i & 31)) & 1u) {
        const unsigned* __restrict__ r = sup + (size_t)i * NMS_WORDS + lane;
#pragma unroll
        for (int t = 0; t < 8; ++t) kw[t] &= ~r[t * 32];
      }
    }
  }

  // Scatter keep bits back to original box order.
#pragma unroll
  for (int t = 0; t < 8; ++t) {
    const int w = t * 32 + lane;  // word index this lane owns
    const unsigned bits = kw[t];
    for (int b = 0; b < 32; ++b) {
      const int i = w * 32 + b;
      if (i < n) out[order[i]] = ((bits >> b) & 1u) ? 1.0f : 0.0f;
    }
  }
}

// ---------------------------------------------------------------------------
extern "C" void kernel_launch(void* const* d_in, const int* in_sizes, int n_in,
                              void* d_out, int out_size, void* d_ws, size_t ws_size,
                              hipStream_t stream) {
  const float4* boxes  = (const float4*)d_in[0];
  const float*  scores = (const float*)d_in[1];
  const int n = in_sizes[1];  // 8192

  char* ws = (char*)d_ws;
  unsigned* order  = (unsigned*)ws;                                  // N u32
  float4*   sboxes = (float4*)(ws + (size_t)n * sizeof(unsigned));   // N f4
  unsigned* sup    = (unsigned*)(ws + (size_t)n * sizeof(unsigned)
                                    + (size_t)n * sizeof(float4));   // N*(N/32)

  nms_sort_kernel<<<1, 1024, 0, stream>>>(boxes, scores, order, sboxes, n);

  dim3 grid(n / 64, n / 64);
  nms_sup_tile_kernel<<<grid, 256, 0, stream>>>(sboxes, sup, n);

  nms_scan_kernel<<<1, 32, 0, stream>>>(sup, order, (float*)d_out, n);
}